// DistanceLoss_14216341749924
// MI455X (gfx1250) — compile-verified
//
#include <hip/hip_runtime.h>
#include <math.h>

typedef __attribute__((ext_vector_type(4))) float f4;
typedef __attribute__((ext_vector_type(4))) int   v4i;
typedef __attribute__((address_space(1))) v4i gv4i;   // global
typedef __attribute__((address_space(3))) v4i lv4i;   // LDS

#define BLOCK 256
#define WAVES_PER_BLOCK 8
#define CMAX 2048   // LDS capacity for centers (2*CMAX floats = 16 KB; C=1000 fits)

#if defined(__has_builtin)
#if __has_builtin(__builtin_amdgcn_global_load_async_to_lds_b128) && \
    __has_builtin(__builtin_amdgcn_s_wait_asynccnt)
#define USE_ASYNC_LDS 1
#endif
#endif

__device__ __forceinline__ void upd(float v, int c, float& best, int& bestIdx) {
    // strictly-greater keeps the first (lowest-index) occurrence within a lane,
    // since a lane visits its columns in increasing order
    if (v > best) { best = v; bestIdx = c; }
}

__global__ __launch_bounds__(BLOCK)
void argmax_dist_kernel(const float* __restrict__ pred,
                        const int*   __restrict__ tru,
                        const float* __restrict__ centers,
                        float*       __restrict__ partials,
                        int B, int C)
{
    __shared__ float ldsC[2 * CMAX];
    __shared__ float waveSum[WAVES_PER_BLOCK];

    const int tid  = threadIdx.x;
    const int lane = tid & 31;
    const int wave = tid >> 5;

    // ---- Stage the centers table (2*C floats) into LDS ----
    const int n = 2 * C;          // 2000 floats for C=1000
    const int nv = n >> 2;        // float4 count (500)
#if USE_ASYNC_LDS
    for (int i = tid; i < nv; i += BLOCK) {
        __builtin_amdgcn_global_load_async_to_lds_b128(
            (gv4i*)(centers + i * 4),
            (lv4i*)(&ldsC[i * 4]),
            0, 0);
    }
    __builtin_amdgcn_s_wait_asynccnt(0);
#else
    for (int i = tid; i < nv; i += BLOCK) {
        *(f4*)(&ldsC[i * 4]) = *(const f4*)(centers + i * 4);
    }
#endif
    if (tid == 0) {                        // scalar remainder (none for C=1000)
        for (int i = nv * 4; i < n; ++i) ldsC[i] = centers[i];
    }
    __syncthreads();

    // ---- One wave per row: streaming argmax over C columns ----
    const int row = blockIdx.x * WAVES_PER_BLOCK + wave;
    float dist = 0.0f;
    if (row < B) {
        const float* rp = pred + (size_t)row * (size_t)C;
        float best = -__builtin_inff();
        int bestIdx = C;

        int col0 = 0;
        for (; col0 + 128 <= C; col0 += 128) {
            const int c = col0 + lane * 4;
            // pred is 262 MB streamed once (> 192 MB L2): non-temporal b128 loads
            f4 v = __builtin_nontemporal_load((const f4*)(rp + c));
            upd(v.x, c + 0, best, bestIdx);
            upd(v.y, c + 1, best, bestIdx);
            upd(v.z, c + 2, best, bestIdx);
            upd(v.w, c + 3, best, bestIdx);
        }
        {   // tail (C=1000: lanes 0..25 take a full, aligned float4)
            const int c = col0 + lane * 4;
            if (c + 3 < C) {
                f4 v = __builtin_nontemporal_load((const f4*)(rp + c));
                upd(v.x, c + 0, best, bestIdx);
                upd(v.y, c + 1, best, bestIdx);
                upd(v.z, c + 2, best, bestIdx);
                upd(v.w, c + 3, best, bestIdx);
            } else {
                for (int k = 0; k < 4; ++k) {
                    if (c + k < C) upd(rp[c + k], c + k, best, bestIdx);
                }
            }
        }

        // ---- wave32 argmax reduction (tie -> lowest index, like jnp.argmax) ----
        #pragma unroll
        for (int off = 16; off > 0; off >>= 1) {
            float ov = __shfl_xor(best, off, 32);
            int   oi = __shfl_xor(bestIdx, off, 32);
            if (ov > best || (ov == best && oi < bestIdx)) { best = ov; bestIdx = oi; }
        }

        // ---- gather both centers from LDS, distance ----
        const float ax = ldsC[2 * bestIdx + 0];
        const float ay = ldsC[2 * bestIdx + 1];
        const int   t  = tru[row];
        const float bx = ldsC[2 * t + 0];
        const float by = ldsC[2 * t + 1];
        const float dx = ax - bx, dy = ay - by;
        dist = sqrtf(dx * dx + dy * dy) * (1.0f / 255.0f);
    }

    if (lane == 0) waveSum[wave] = dist;
    __syncthreads();
    if (tid == 0) {
        float s = 0.0f;
        #pragma unroll
        for (int w = 0; w < WAVES_PER_BLOCK; ++w) s += waveSum[w];
        partials[blockIdx.x] = s;
    }
}

// Deterministic single-block final reduction (fixed accumulation order).
__global__ __launch_bounds__(BLOCK)
void reduce_kernel(const float* __restrict__ partials, int n,
                   float* __restrict__ out)
{
    __shared__ float s[BLOCK];
    const int tid = threadIdx.x;
    float acc = 0.0f;
    for (int i = tid; i < n; i += BLOCK) acc += partials[i];
    s[tid] = acc;
    __syncthreads();
    #pragma unroll
    for (int stride = BLOCK / 2; stride > 0; stride >>= 1) {
        if (tid < stride) s[tid] += s[tid + stride];
        __syncthreads();
    }
    if (tid == 0) out[0] = s[0];
}

extern "C" void kernel_launch(void* const* d_in, const int* in_sizes, int n_in,
                              void* d_out, int out_size, void* d_ws, size_t ws_size,
                              hipStream_t stream)
{
    const float* pred    = (const float*)d_in[0];   // [B, C] f32
    const int*   tru     = (const int*)d_in[1];     // [B] int
    const float* centers = (const float*)d_in[2];   // [C, 2] f32

    const int B = in_sizes[1];
    const int C = in_sizes[2] / 2;

    float* partials = (float*)d_ws;
    const int blocks = (B + WAVES_PER_BLOCK - 1) / WAVES_PER_BLOCK;   // 8192

    argmax_dist_kernel<<<blocks, BLOCK, 0, stream>>>(pred, tru, centers,
                                                     partials, B, C);
    reduce_kernel<<<1, BLOCK, 0, stream>>>(partials, blocks, (float*)d_out);
}